// TransformerDecoder_68358699483655
// MI455X (gfx1250) — compile-verified
//
#include <hip/hip_runtime.h>

typedef __attribute__((ext_vector_type(2))) float v2f;
typedef __attribute__((ext_vector_type(8))) float v8f;

#define D_MODEL 768
#define SEQ_W   512
#define VOCAB   10000
#define HID_N   3072
#define NBLK    4
#define NHEAD   12
#define DKH     64
#define LOGITS_LD 10016   // 10000 padded to multiple of 16

// ---------------- position vector: posvec = pids @ Wp + bp ----------------
__global__ void pos_kernel(const float* __restrict__ pids, const float* __restrict__ Wp,
                           const float* __restrict__ bp, float* __restrict__ posvec) {
  int d = blockIdx.x * blockDim.x + threadIdx.x;
  if (d >= D_MODEL) return;
  float s = bp[d];
  for (int w = 0; w < SEQ_W; ++w) s += pids[w] * Wp[w * D_MODEL + d];
  posvec[d] = s;
}

// ------- last-position one-hot gather + pos add; zero pad rows 8..15 -------
__global__ void embed_kernel(const float* __restrict__ X, const float* __restrict__ We,
                             const float* __restrict__ be, const float* __restrict__ posvec,
                             float* __restrict__ x) {
  int r = blockIdx.x;
  int tid = threadIdx.x;
  if (r >= 8) {  // zero padding rows so WMMA M=16 pad contributes nothing
    for (int d = tid; d < D_MODEL; d += blockDim.x) x[r * D_MODEL + d] = 0.0f;
    return;
  }
  __shared__ int sidx;
  if (tid == 0) sidx = 0;
  __syncthreads();
  const float* row = X + ((size_t)r * SEQ_W + (SEQ_W - 1)) * (size_t)VOCAB;
  for (int v = tid; v < VOCAB; v += blockDim.x)
    if (row[v] > 0.5f) sidx = v;     // exactly one hot element -> single writer
  __syncthreads();
  int idx = sidx;
  for (int d = tid; d < D_MODEL; d += blockDim.x)
    x[r * D_MODEL + d] = We[(size_t)idx * D_MODEL + d] + be[d] + posvec[d];
}

// ---------------- Y[16,N] = A[16,K] @ B (+bias)(+relu) via WMMA f32 16x16x4 ----------------
// MODE 0: B row-major [K x ldb].  MODE 1: Wv packed [H][D][DK] -> eff B[d, h*64+k].
// GUARD: per-column bounds check (only the Wl tail block needs it).
template <int MODE, bool GUARD>
__global__ __launch_bounds__(128)
void gemm16_wmma(const float* __restrict__ A, const float* __restrict__ B,
                 const float* __restrict__ bias, float* __restrict__ Y,
                 int K, int N, int ldb, int ldy, int do_relu, int tile_base) {
  const int lane = threadIdx.x & 31;
  const int half = lane >> 4;          // 0: K pair {k,k+1}; 1: {k+2,k+3}
  const int l16  = lane & 15;
  const int waveId = blockIdx.x * (blockDim.x >> 5) + (threadIdx.x >> 5);
  const int tile0  = tile_base + waveId * 4;   // 4 N-tiles of 16 per wave

  int  col[4];
  bool ok[4];
#pragma unroll
  for (int t = 0; t < 4; ++t) {
    col[t] = (tile0 + t) * 16 + l16;
    ok[t]  = GUARD ? (col[t] < N) : true;
  }

  v8f acc[4] = {};
  const float* Arow = A + (size_t)l16 * K;
  const int half2 = half * 2;

  for (int k = 0; k < K; k += 4) {
    const int ka = k + half2;
    const v2f a = *reinterpret_cast<const v2f*>(Arow + ka);  // 8B-aligned: K%4==0, ka even

    float bx[4], by[4];
#pragma unroll
    for (int t = 0; t < 4; ++t) {
      if (GUARD && !ok[t]) { bx[t] = 0.0f; by[t] = 0.0f; }
      else if (MODE == 0) {
        const float* p = B + (size_t)ka * ldb + col[t];
        bx[t] = p[0];
        by[t] = p[ldb];
      } else {
        // Wv element (d=ka, col): (col>>6)*D*64 + d*64 + (col&63)
        const float* p = B + ((size_t)(col[t] >> 6) * D_MODEL + ka) * DKH + (col[t] & (DKH - 1));
        bx[t] = p[0];
        by[t] = p[DKH];
      }
    }
#pragma unroll
    for (int t = 0; t < 4; ++t) {
      v2f b; b.x = bx[t]; b.y = by[t];
      acc[t] = __builtin_amdgcn_wmma_f32_16x16x4_f32(
          /*neg_a=*/false, a, /*neg_b=*/false, b,
          /*c_mod=*/(short)0, acc[t], /*reuse_a=*/false, /*reuse_b=*/false);
    }
    // prefetch the B stream a few K-steps ahead (global_prefetch_b8)
    if (MODE == 0 && !GUARD && (k + 32) < K)
      __builtin_prefetch(B + (size_t)(k + 32 + half2) * ldb + col[0], 0, 1);
  }

#pragma unroll
  for (int t = 0; t < 4; ++t) {
    if (!GUARD || ok[t]) {
      const float bb = bias[col[t]];
#pragma unroll
      for (int j = 0; j < 8; ++j) {
        const int row = j + half * 8;   // C/D layout: VGPR j -> M=j (lanes 0-15), M=j+8 (lanes 16-31)
        float v = acc[t][j] + bb;
        if (do_relu) v = fmaxf(v, 0.0f);
        Y[(size_t)row * ldy + col[t]] = v;
      }
    }
  }
}

// ---------------- x[r,:] = LayerNorm(x[r,:] + o[r,:])  (biased var, no eps) --------------
__global__ void add_ln_kernel(float* __restrict__ x, const float* __restrict__ o) {
  const int r = blockIdx.x;      // rows 0..7 only; pad rows of x stay zero
  const int tid = threadIdx.x;   // 256 threads, 3 elems each
  __shared__ float ssum[256];
  __shared__ float ssq[256];
  float vals[3];
  float ls = 0.0f, lq = 0.0f;
#pragma unroll
  for (int j = 0; j < 3; ++j) {
    const int d = tid + j * 256;
    const float v = x[r * D_MODEL + d] + o[r * D_MODEL + d];
    vals[j] = v; ls += v; lq += v * v;
  }
  ssum[tid] = ls; ssq[tid] = lq;
  __syncthreads();
  for (int s = 128; s > 0; s >>= 1) {
    if (tid < s) { ssum[tid] += ssum[tid + s]; ssq[tid] += ssq[tid + s]; }
    __syncthreads();
  }
  const float mu  = ssum[0] * (1.0f / D_MODEL);
  const float var = ssq[0] * (1.0f / D_MODEL) - mu * mu;
  const float inv = rsqrtf(var);
#pragma unroll
  for (int j = 0; j < 3; ++j) {
    const int d = tid + j * 256;
    x[r * D_MODEL + d] = (vals[j] - mu) * inv;
  }
}

// ---------------- row softmax over 10000 logits -> out[8,10000] ----------------
__global__ void softmax_kernel(const float* __restrict__ L, float* __restrict__ out) {
  const int r = blockIdx.x;
  const int tid = threadIdx.x;
  __shared__ float red[256];
  const float* row = L + (size_t)r * LOGITS_LD;
  float m = -3.402823466e38f;
  for (int v = tid; v < VOCAB; v += 256) m = fmaxf(m, row[v]);
  red[tid] = m; __syncthreads();
  for (int s = 128; s > 0; s >>= 1) { if (tid < s) red[tid] = fmaxf(red[tid], red[tid + s]); __syncthreads(); }
  const float mx = red[0];
  __syncthreads();
  float sum = 0.0f;
  for (int v = tid; v < VOCAB; v += 256) sum += expf(row[v] - mx);
  red[tid] = sum; __syncthreads();
  for (int s = 128; s > 0; s >>= 1) { if (tid < s) red[tid] += red[tid + s]; __syncthreads(); }
  const float inv = 1.0f / red[0];
  for (int v = tid; v < VOCAB; v += 256)
    out[(size_t)r * VOCAB + v] = expf(row[v] - mx) * inv;
}

extern "C" void kernel_launch(void* const* d_in, const int* in_sizes, int n_in,
                              void* d_out, int out_size, void* d_ws, size_t ws_size,
                              hipStream_t stream) {
  const float* X    = (const float*)d_in[0];
  const float* pids = (const float*)d_in[1];
  const float* We   = (const float*)d_in[2];
  const float* be   = (const float*)d_in[3];
  const float* Wp   = (const float*)d_in[4];
  const float* bp   = (const float*)d_in[5];
  // d_in[6..9] = Wq,bq,Wk,bk -> dead code in the reference
  const float* Wv   = (const float*)d_in[10];
  const float* bv   = (const float*)d_in[11];
  const float* Wo   = (const float*)d_in[12];
  const float* bo   = (const float*)d_in[13];
  const float* W1   = (const float*)d_in[14];
  const float* b1   = (const float*)d_in[15];
  const float* W2   = (const float*)d_in[16];
  const float* b2   = (const float*)d_in[17];
  const float* Wl   = (const float*)d_in[18];
  const float* bl   = (const float*)d_in[19];
  float* out = (float*)d_out;

  float* ws     = (float*)d_ws;
  float* posvec = ws;                      //   768
  float* x      = posvec + D_MODEL;        // 16*768
  float* y      = x + 16 * D_MODEL;        // 16*768
  float* o      = y + 16 * D_MODEL;        // 16*768
  float* f      = o + 16 * D_MODEL;        // 16*3072
  float* logits = f + 16 * HID_N;          // 16*10016

  pos_kernel<<<3, 256, 0, stream>>>(pids, Wp, bp, posvec);
  embed_kernel<<<16, 256, 0, stream>>>(X, We, be, posvec, x);

  // MODE-0 GEMM: full 256-col blocks run unguarded; remainder (Wl only) runs guarded.
  auto gemm0 = [&](const float* A, const float* B, const float* bias, float* Y,
                   int K, int N, int ldb, int ldy, int relu) {
    int fullBlocks = N / 256;              // 128 thr = 4 waves * 64 cols = 256 cols/block
    if (fullBlocks > 0)
      gemm16_wmma<0, false><<<fullBlocks, 128, 0, stream>>>(A, B, bias, Y, K, N, ldb, ldy, relu, 0);
    if (fullBlocks * 256 < N)
      gemm16_wmma<0, true><<<1, 128, 0, stream>>>(A, B, bias, Y, K, N, ldb, ldy, relu, fullBlocks * 16);
  };

  for (int i = 0; i < NBLK; ++i) {
    // "attention" (degenerate: A == concat of per-head V); N=768 -> 3 full blocks
    gemm16_wmma<1, false><<<3, 128, 0, stream>>>(
        x, Wv + (size_t)i * NHEAD * D_MODEL * DKH, bv + i * NHEAD * DKH,
        y, D_MODEL, D_MODEL, 0, D_MODEL, 0, 0);
    gemm0(y, Wo + (size_t)i * D_MODEL * D_MODEL, bo + i * D_MODEL,
          o, D_MODEL, D_MODEL, D_MODEL, D_MODEL, 0);
    add_ln_kernel<<<8, 256, 0, stream>>>(x, o);
    // FFN
    gemm0(x, W1 + (size_t)i * D_MODEL * HID_N, b1 + i * HID_N,
          f, D_MODEL, HID_N, HID_N, HID_N, /*relu=*/1);
    gemm0(f, W2 + (size_t)i * HID_N * D_MODEL, b2 + i * D_MODEL,
          o, HID_N, D_MODEL, D_MODEL, D_MODEL, 0);
    add_ln_kernel<<<8, 256, 0, stream>>>(x, o);
  }

  // final projection: only last position matters -> [8(padded 16), 768] x [768, 10000]
  gemm0(x, Wl, bl, logits, D_MODEL, VOCAB, VOCAB, LOGITS_LD, 0);
  softmax_kernel<<<8, 256, 0, stream>>>(logits, out);
}